// DynamicGCN_43267500540702
// MI455X (gfx1250) — compile-verified
//
#include <hip/hip_runtime.h>
#include <hip/hip_bf16.h>

typedef __attribute__((ext_vector_type(2))) float v2f;
typedef __attribute__((ext_vector_type(8))) float v8f;

#define N_NODES_C 100000
#define N_EDGES_C 600000
#define HID 128
#define PSTR 288   // floats per K-pair row in sWp; 288 % 64 == 32 -> conflict-free ds_load_b64

// ---------------- temporal gate: sigmoid(tanh(t*Wg1+bg1) @ Wg2 + bg2) ----------------
__global__ void gate_kernel(const float* __restrict__ ts,
                            const float* __restrict__ Wg1, const float* __restrict__ bg1,
                            const float* __restrict__ Wg2, const float* __restrict__ bg2,
                            float* __restrict__ gate) {
  __shared__ float hid[HID];
  int j = threadIdx.x;
  float t = ts[0];
  hid[j] = tanhf(t * Wg1[j] + bg1[j]);
  __syncthreads();
  float acc = bg2[j];
  for (int k = 0; k < HID; ++k) acc += hid[k] * Wg2[k * HID + j];
  gate[j] = 1.0f / (1.0f + expf(-acc));
}

__global__ void zero_kernel(float* __restrict__ p, int n) {
  int i = blockIdx.x * blockDim.x + threadIdx.x;
  if (i < n) p[i] = 0.0f;
}

__global__ void deg_scatter_kernel(const int* __restrict__ dst, float* __restrict__ deg, int ne) {
  int e = blockIdx.x * blockDim.x + threadIdx.x;
  if (e < ne) unsafeAtomicAdd(&deg[dst[e]], 1.0f);
}

__global__ void deg_fin_kernel(const float* __restrict__ deg,
                               float* __restrict__ dinv, float* __restrict__ winv, int n) {
  int i = blockIdx.x * blockDim.x + threadIdx.x;
  if (i < n) {
    float d = deg[i] + 1.0f;       // +1 for self-loop
    dinv[i] = rsqrtf(d);
    winv[i] = 1.0f / d;
  }
}

// ---------------- fused GEMM: H = X @ W ; ACC = H*winv[row] + bias[col] ----------------
// block = 256 threads (8 waves). Each block computes 128 rows x 128 cols.
// Each wave computes a 16x128 strip via 8 accumulator tiles of V_WMMA_F32_16X16X4_F32.
__global__ __launch_bounds__(256)
void gemm_gcn_kernel(const float* __restrict__ X, const float* __restrict__ W,
                     const float* __restrict__ winv, const float* __restrict__ bias,
                     float* __restrict__ H, float* __restrict__ ACC, int nrows) {
  // W staged K-pair-interleaved: sWp[p*PSTR + 2n + 0/1] = W[2p][n], W[2p+1][n]
  // -> each B fragment is ONE aligned ds_load_b64 into an even VGPR pair (no v_mov fixups)
  __shared__ float sWp[64 * PSTR];     // 73.7 KB
  __shared__ float sX[128 * 132];      // 67.6 KB, [m][k], pitch 132: banks (4m+k)%64, conflict-free

  const int tid = threadIdx.x;
  const int r0  = blockIdx.x * 128;

  // stage W with on-the-fly pair interleave
  for (int i = tid; i < 64 * 32; i += 256) {
    int p  = i >> 5;                   // k-pair index, k = 2p
    int n4 = (i & 31) * 4;             // starting column
    float4 ra = *(const float4*)(W + (size_t)(2 * p)     * HID + n4);
    float4 rb = *(const float4*)(W + (size_t)(2 * p + 1) * HID + n4);
    v2f* dp = (v2f*)(sWp + p * PSTR + n4 * 2);   // byte addr p*1152 + n4*8 : 8B aligned
    v2f t0; t0.x = ra.x; t0.y = rb.x; dp[0] = t0;
    v2f t1; t1.x = ra.y; t1.y = rb.y; dp[1] = t1;
    v2f t2; t2.x = ra.z; t2.y = rb.z; dp[2] = t2;
    v2f t3; t3.x = ra.w; t3.y = rb.w; dp[3] = t3;
  }

  // stage the 128-row X tile, coalesced float4 loads; OOB rows -> 0
  for (int i = tid; i < 128 * (HID / 4); i += 256) {
    int row = i >> 5;                  // 32 float4 per row
    int c4  = i & 31;
    int g   = r0 + row;
    float4 v = make_float4(0.f, 0.f, 0.f, 0.f);
    if (g < nrows) v = *(const float4*)(X + (size_t)g * HID + c4 * 4);
    *(float4*)(sX + row * 132 + c4 * 4) = v;     // pitch 528B, 16B-aligned
  }
  __syncthreads();

  const int wave = tid >> 5;
  const int lane = tid & 31;
  const int lo   = lane & 15;
  const int hi   = lane >> 4;
  const int m0   = wave * 16;          // wave's 16-row strip inside the tile

  v8f c[8] = {};                       // 8 N-tiles of 16x16 f32 accumulators (64 VGPRs)

  for (int k0 = 0; k0 < HID; k0 += 4) {
    // A 16x4 fragment: lanes 0-15 hold K=k0,k0+1 ; lanes 16-31 hold K=k0+2,k0+3
    v2f a = *(const v2f*)(sX + (m0 + lo) * 132 + k0 + 2 * hi);
    const int p = (k0 >> 1) + hi;      // K-pair index for this half-wave
#pragma unroll
    for (int t = 0; t < 8; ++t) {
      v2f b = *(const v2f*)(sWp + p * PSTR + (t * 16 + lo) * 2);  // one ds_load_b64
      c[t] = __builtin_amdgcn_wmma_f32_16x16x4_f32(
          /*neg_a=*/false, a, /*neg_b=*/false, b,
          /*c_mod=*/(short)0, c[t], /*reuse_a=*/false, /*reuse_b=*/false);
    }
  }

  // epilogue: C layout — VGPR v: lanes0-15 row m0+v, lanes16-31 row m0+v+8; col = t*16+lo
  float wr[8];
#pragma unroll
  for (int v = 0; v < 8; ++v) {
    int row = r0 + m0 + v + 8 * hi;
    wr[v] = (row < nrows) ? winv[row] : 0.0f;
  }
#pragma unroll
  for (int t = 0; t < 8; ++t) {
    int n  = t * 16 + lo;
    float bn = bias[n];
#pragma unroll
    for (int v = 0; v < 8; ++v) {
      int row = r0 + m0 + v + 8 * hi;
      if (row < nrows) {
        float hv = c[t][v];
        size_t off = (size_t)row * HID + n;
        H[off]   = hv;
        ACC[off] = hv * wr[v] + bn;    // self-loop term + bias, pre-initialized for scatter
      }
    }
  }
}

// ---------------- edge scatter: ACC[dst] += H[src] * dinv[src]*dinv[dst] ----------------
// one wave32 per edge: e is wave-uniform (indices scalarize), each lane handles 4 features
__global__ void edge_scatter_kernel(const int* __restrict__ src, const int* __restrict__ dst,
                                    const float* __restrict__ H, const float* __restrict__ dinv,
                                    float* __restrict__ ACC, int ne) {
  int t = blockIdx.x * blockDim.x + threadIdx.x;
  int e = t >> 5;
  int lane = t & 31;
  if (e < ne) {
    int s = src[e];
    int d = dst[e];
    float w = dinv[s] * dinv[d];
    float4 h4 = *(const float4*)(H + (size_t)s * HID + lane * 4);
    float* a = ACC + (size_t)d * HID + lane * 4;
    unsafeAtomicAdd(a + 0, h4.x * w);
    unsafeAtomicAdd(a + 1, h4.y * w);
    unsafeAtomicAdd(a + 2, h4.z * w);
    unsafeAtomicAdd(a + 3, h4.w * w);
  }
}

// ---------------- finalize: out = relu(acc) * gate[col] ----------------
__global__ void finalize_kernel(const float* __restrict__ ACC, const float* __restrict__ gate,
                                float* __restrict__ OUT, int n) {
  int i = blockIdx.x * blockDim.x + threadIdx.x;
  if (i < n) {
    float v = ACC[i];
    OUT[i] = fmaxf(v, 0.0f) * gate[i & (HID - 1)];
  }
}

extern "C" void kernel_launch(void* const* d_in, const int* in_sizes, int n_in,
                              void* d_out, int out_size, void* d_ws, size_t ws_size,
                              hipStream_t stream) {
  (void)in_sizes; (void)n_in; (void)out_size; (void)ws_size;

  const float* x   = (const float*)d_in[0];
  const int*   ei  = (const int*)d_in[1];    // edge_index [2, E]; harness passes integers as int32
  const float* ts  = (const float*)d_in[2];
  const float* Wl[3] = {(const float*)d_in[3], (const float*)d_in[5], (const float*)d_in[7]};
  const float* bl[3] = {(const float*)d_in[4], (const float*)d_in[6], (const float*)d_in[8]};
  const float* Wg1 = (const float*)d_in[9];
  const float* bg1 = (const float*)d_in[10];
  const float* Wg2 = (const float*)d_in[11];
  const float* bg2 = (const float*)d_in[12];

  const int* src = ei;
  const int* dst = ei + N_EDGES_C;

  // workspace layout
  float* bufA = (float*)d_ws;                              // H buffer  (51.2 MB)
  float* bufB = bufA + (size_t)N_NODES_C * HID;            // ACC / activations (51.2 MB)
  float* deg  = bufB + (size_t)N_NODES_C * HID;            // 400 KB
  float* dinv = deg  + N_NODES_C;
  float* winv = dinv + N_NODES_C;
  float* gate = winv + N_NODES_C;                          // 512 B

  // one-time prep (deterministic every call)
  gate_kernel<<<1, HID, 0, stream>>>(ts, Wg1, bg1, Wg2, bg2, gate);
  zero_kernel<<<(N_NODES_C + 255) / 256, 256, 0, stream>>>(deg, N_NODES_C);
  deg_scatter_kernel<<<(N_EDGES_C + 255) / 256, 256, 0, stream>>>(dst, deg, N_EDGES_C);
  deg_fin_kernel<<<(N_NODES_C + 255) / 256, 256, 0, stream>>>(deg, dinv, winv, N_NODES_C);

  float* out = (float*)d_out;
  const float* xin = x;
  const int nfeat = N_NODES_C * HID;

  for (int l = 0; l < 3; ++l) {
    float* acc = (l == 2) ? out : bufB;
    // GEMM reads its X rows into LDS before overwriting ACC rows, so acc==xin is safe per-block
    gemm_gcn_kernel<<<(N_NODES_C + 127) / 128, 256, 0, stream>>>(
        xin, Wl[l], winv, bl[l], bufA, acc, N_NODES_C);
    edge_scatter_kernel<<<(int)(((size_t)N_EDGES_C * 32 + 255) / 256), 256, 0, stream>>>(
        src, dst, bufA, dinv, acc, N_EDGES_C);
    finalize_kernel<<<(nfeat + 255) / 256, 256, 0, stream>>>(acc, gate, acc, nfeat);
    xin = acc;
  }
}